// JointLearner_19705309954583
// MI455X (gfx1250) — compile-verified
//
#include <hip/hip_runtime.h>

typedef float v2f __attribute__((ext_vector_type(2)));
typedef float v8f __attribute__((ext_vector_type(8)));

#define FDIM 32
#define HDIM 512
#define LDS_A_STRIDE 34   // pad 16x32 A tile rows to dodge LDS bank conflicts

// ---------------------------------------------------------------------------
// Kernel 1: deterministic ragged segment-sum.
// One wave32 per segment; lane = feature index (F==32 == wave width).
// Segments of a token are contiguous in seg_token_idx (np.repeat of distinct
// ranks), so the wave owning the FIRST segment of a run accumulates the run
// (length <= 3) and writes the token row exactly once. No atomics => the
// result is bit-identical on every replay.
// ---------------------------------------------------------------------------
__global__ void __launch_bounds__(256)
seg_sum_kernel(const float* __restrict__ features,
               const int*   __restrict__ seg_idx,
               float*       __restrict__ tokens,
               int S) {
    const int s    = (blockIdx.x * blockDim.x + threadIdx.x) >> 5;  // segment id
    const int lane = threadIdx.x & 31;                              // feature
    if (s >= S) return;
    const int tok = seg_idx[s];
    const bool first = (s == 0) || (seg_idx[s - 1] != tok);
    if (!first) return;                 // uniform per wave (whole wave shares s)
    // features is a 34MB read-once stream: non-temporal loads.
    float acc = __builtin_nontemporal_load(&features[s * FDIM + lane]);
    int j = s + 1;
    while (j < S && seg_idx[j] == tok) {        // at most 2 extra iterations
        acc += __builtin_nontemporal_load(&features[j * FDIM + lane]);
        ++j;
    }
    // tokens (16.9MB) is re-read by the GEMM: keep regular-temporal store.
    tokens[tok * FDIM + lane] = acc + 1e-10f;
}

// ---------------------------------------------------------------------------
// Kernel 2: out[T,512] = tokens[T,32] @ W[32,512] + b via V_WMMA_F32_16X16X4_F32.
// Block = 16 output rows (one M-tile), 8 waves; wave w owns columns [w*64, w*64+64).
// K=32 => 8 chained WMMAs per 16x16 tile, C seeded with the column bias.
// ---------------------------------------------------------------------------
__global__ void __launch_bounds__(256)
gemm_wmma_kernel(const float* __restrict__ tokens,
                 const float* __restrict__ W,
                 const float* __restrict__ bias,
                 float*       __restrict__ out,
                 int T) {
    __shared__ float ldsA[16 * LDS_A_STRIDE];

    const int m0  = blockIdx.x * 16;
    const int tid = threadIdx.x;

    // Stage the 16x32 A tile into LDS: each of the 256 threads moves one v2f.
    {
        const int r = tid >> 4;              // 0..15 (tile row)
        const int c = (tid & 15) * 2;        // 0,2,...,30 (k column)
        int gr = m0 + r;
        if (gr >= T) gr = T - 1;             // safe clamp for a partial last tile
        v2f a = *(const v2f*)(tokens + gr * FDIM + c);
        *(v2f*)(&ldsA[r * LDS_A_STRIDE + c]) = a;
    }
    __syncthreads();

    const int wave = tid >> 5;
    const int lane = tid & 31;
    const int half = lane >> 4;              // 0: lanes 0-15, 1: lanes 16-31
    const int mr   = lane & 15;              // A row within tile / D column idx

    // A fragments for all 8 K-steps (ISA 16x4 f32 layout):
    //   vgpr0 = A[mr][k0 + 2*half], vgpr1 = A[mr][k0 + 2*half + 1]
    v2f afrag[8];
#pragma unroll
    for (int ks = 0; ks < 8; ++ks) {
        afrag[ks] = *(const v2f*)(&ldsA[mr * LDS_A_STRIDE + ks * 4 + 2 * half]);
    }

    const bool fullTile = (m0 + 16 <= T);    // uniform: one scalar branch/block
    const int n0 = wave * 64;
#pragma unroll
    for (int t = 0; t < 4; ++t) {
        const int n  = n0 + t * 16 + mr;     // output column for this lane
        const float bv = bias[n];
        v8f c = {bv, bv, bv, bv, bv, bv, bv, bv};   // C = broadcast bias
#pragma unroll
        for (int ks = 0; ks < 8; ++ks) {
            const int krow = ks * 4 + 2 * half;
            v2f bfrag;
            bfrag.x = W[krow * HDIM + n];            // B vgpr0 = W[k][n]
            bfrag.y = W[(krow + 1) * HDIM + n];      // B vgpr1 = W[k+1][n]
            c = __builtin_amdgcn_wmma_f32_16x16x4_f32(
                    /*neg_a=*/false, afrag[ks],
                    /*neg_b=*/false, bfrag,
                    /*c_mod=*/(short)0, c,
                    /*reuse_a=*/false, /*reuse_b=*/false);
        }
        // D layout: vgpr i -> row (i + 8*half), lane%16 -> column.
        // Output (270MB) is written once, never re-read: non-temporal stores
        // so it streams past L2. Hot path = 8 straight unguarded stores.
        if (fullTile) {
#pragma unroll
            for (int i = 0; i < 8; ++i) {
                __builtin_nontemporal_store(c[i], &out[(m0 + i + 8 * half) * HDIM + n]);
            }
        } else {
#pragma unroll
            for (int i = 0; i < 8; ++i) {
                const int row = m0 + i + 8 * half;
                if (row < T) __builtin_nontemporal_store(c[i], &out[row * HDIM + n]);
            }
        }
    }
}

// ---------------------------------------------------------------------------
extern "C" void kernel_launch(void* const* d_in, const int* in_sizes, int n_in,
                              void* d_out, int out_size, void* d_ws, size_t ws_size,
                              hipStream_t stream) {
    const float* features = (const float*)d_in[0];   // [S, 32] f32
    const float* W        = (const float*)d_in[1];   // [32, 512] f32
    const float* b        = (const float*)d_in[2];   // [512] f32
    const int*   seg_idx  = (const int*)d_in[3];     // [S] i32
    // d_in[4] = num_tokens (device-resident scalar; host derives T below)

    const int S = in_sizes[0] / FDIM;
    const int T = out_size / HDIM;

    float* tokens = (float*)d_ws;                    // [T, 32] scratch (16.9 MB)

    const int segBlocks = (S + 7) / 8;               // 8 waves (segments) / block
    seg_sum_kernel<<<segBlocks, 256, 0, stream>>>(features, seg_idx, tokens, S);

    const int mTiles = (T + 15) / 16;
    gemm_wmma_kernel<<<mTiles, 256, 0, stream>>>(tokens, W, b, (float*)d_out, T);
}